// NegativeSample_46471546142940
// MI455X (gfx1250) — compile-verified
//
#include <hip/hip_runtime.h>
#include <math.h>

typedef __attribute__((ext_vector_type(2))) float v2f;
typedef __attribute__((ext_vector_type(4))) float v4f;
typedef __attribute__((ext_vector_type(8))) float v8f;

#define TILE    128
#define KDIM    128
#define LSTRIDE 132   // 128 + 4 pad: bank = (4*row + kk) & 63 -> conflict-free frag gathers

__global__ void nsl_zero_out(float* out) { out[0] = 0.0f; }

__launch_bounds__(256)
__global__ void nsl_wmma_kernel(const float* __restrict__ E,
                                float* __restrict__ out, int n) {
    // Triangular block decode: idx -> (bi, bj) with bi >= bj.
    const int idx = blockIdx.x;
    int bi = (int)((sqrtf(8.0f * (float)idx + 1.0f) - 1.0f) * 0.5f);
    while ((bi + 1) * (bi + 2) / 2 <= idx) ++bi;   // fix float rounding
    while (bi * (bi + 1) / 2 > idx) --bi;
    const int bj = idx - bi * (bi + 1) / 2;

    __shared__ float lA[TILE * LSTRIDE];
    __shared__ float lB[TILE * LSTRIDE];
    __shared__ float red[8];

    const int tid     = threadIdx.x;
    const int rowBase = bi * TILE;
    const int colBase = bj * TILE;

    // Cooperative staging: 128 rows x 128 f32 per strip, float4 coalesced.
    for (int i = tid; i < TILE * (KDIM / 4); i += 256) {
        const int r = i >> 5;            // row within strip
        const int c = (i & 31) << 2;     // column (float4)
        v4f a = *(const v4f*)(E + (size_t)(rowBase + r) * KDIM + c);
        *(v4f*)(lA + r * LSTRIDE + c) = a;
        v4f b = *(const v4f*)(E + (size_t)(colBase + r) * KDIM + c);
        *(v4f*)(lB + r * LSTRIDE + c) = b;
    }
    __syncthreads();

    const int wave = tid >> 5;
    const int lane = tid & 31;
    const int l15  = lane & 15;
    const int hk   = (lane >> 4) << 1;   // lanes 16-31 hold K+2,K+3

    v8f acc[8] = {};                      // wave's 16x128 strip: 8 subtiles of 16x16

    const float* __restrict__ Arow = lA + (wave * 16 + l15) * LSTRIDE;

    for (int k = 0; k < KDIM; k += 4) {
        const int kk = k + hk;
        const v2f a = *(const v2f*)(Arow + kk);          // A 16x4 fragment
        #pragma unroll
        for (int nidx = 0; nidx < 8; ++nidx) {
            // B = E^T: B[kk][col] = E[col][kk] -> same gather on column strip
            const v2f b = *(const v2f*)(lB + (nidx * 16 + l15) * LSTRIDE + kk);
            acc[nidx] = __builtin_amdgcn_wmma_f32_16x16x4_f32(
                false, a, false, b, (short)0, acc[nidx], false, false);
        }
    }

    // Epilogue. C/D layout: VGPR v -> M = v (+8 for lanes 16-31); N = lane&15.
    //   Uniform part:  ls(-s) = -max(s,0) - log(1 + exp(-|s|))   for every element
    //   Diagonal fix:  desired ls(s) - ln2 = ls(-s) + (s - ln2)   [ls(x)-ls(-x)=x]
    const float LN2  = 0.69314718055994530942f;
    const bool  diagBlock = (bi == bj);
    const float wgt  = diagBlock ? 1.0f : 2.0f;  // off-diag block stands in for its transpose
    const int   hi8  = (lane >> 4) << 3;

    float lsum = 0.0f;   // sum of ls(-s) over all 64 elements
    float csum = 0.0f;   // diagonal corrections (s - ln2)
    #pragma unroll
    for (int nidx = 0; nidx < 8; ++nidx) {
        #pragma unroll
        for (int v = 0; v < 8; ++v) {
            const float s = acc[nidx][v];
            lsum += -fmaxf(s, 0.0f) - __logf(1.0f + __expf(-fabsf(s)));
        }
        if (diagBlock && nidx == wave) {         // wave-uniform scalar branch
            #pragma unroll
            for (int v = 0; v < 8; ++v) {
                const bool d = (l15 == v + hi8); // this lane holds diag element (M==N)
                csum += d ? (acc[nidx][v] - LN2) : 0.0f;
            }
        }
    }
    float sum = wgt * lsum + csum;

    // wave32 tree reduction, then cross-wave via LDS.
    #pragma unroll
    for (int off = 16; off > 0; off >>= 1)
        sum += __shfl_xor(sum, off, 32);
    if (lane == 0) red[wave] = sum;
    __syncthreads();
    if (tid == 0) {
        float t = 0.0f;
        #pragma unroll
        for (int w = 0; w < 8; ++w) t += red[w];
        atomicAdd(out, -t / (float)n);
    }
}

extern "C" void kernel_launch(void* const* d_in, const int* in_sizes, int n_in,
                              void* d_out, int out_size, void* d_ws, size_t ws_size,
                              hipStream_t stream) {
    (void)n_in; (void)out_size; (void)d_ws; (void)ws_size;
    const float* E = (const float*)d_in[0];
    float* out = (float*)d_out;
    const int n = in_sizes[0] / KDIM;     // 16384
    const int T = n / TILE;               // 128 tile rows/cols
    const int nblocks = T * (T + 1) / 2;  // lower triangle incl. diagonal

    nsl_zero_out<<<dim3(1), dim3(1), 0, stream>>>(out);
    nsl_wmma_kernel<<<dim3(nblocks), dim3(256), 0, stream>>>(E, out, n);
}